// HybridModel_45071386804753
// MI455X (gfx1250) — compile-verified
//
#include <hip/hip_runtime.h>
#include <math.h>

typedef float v2f __attribute__((ext_vector_type(2)));
typedef float v8f __attribute__((ext_vector_type(8)));

#define NQ 4

// Pack W1[4][D] into WMMA-B layout: Wb[kk][n][j] = (n<4 ? W1[n][2*kk+j] : 0)
// float index = kk*32 + n*2 + j   (kk = k/2, j = k&1, n = 0..15)
__global__ void prep_wb_kernel(const float* __restrict__ W1,
                               float* __restrict__ Wb, int D) {
    int idx = blockIdx.x * blockDim.x + threadIdx.x;   // D*16 elements total
    if (idx >= D * 16) return;
    int j  = idx & 1;
    int n  = (idx >> 1) & 15;
    int kk = idx >> 5;
    int k  = 2 * kk + j;
    Wb[idx] = (n < 4) ? W1[n * D + k] : 0.0f;
}

// RX(theta) on wire with index-bit BIT (wire w -> BIT = 1<<(3-w)):
// a0' = c*a0 - i*s*a1 ; a1' = -i*s*a0 + c*a1   (c=cos(t/2), s=sin(t/2))
template<int BIT>
__device__ __forceinline__ void rx_gate(float re[16], float im[16], float c, float s) {
#pragma unroll
    for (int i = 0; i < 16; ++i) {
        if (!(i & BIT)) {
            const int j = i | BIT;
            float r0 = re[i], i0 = im[i], r1 = re[j], i1 = im[j];
            re[i] = c * r0 + s * i1;
            im[i] = c * i0 - s * r1;
            re[j] = c * r1 + s * i0;
            im[j] = c * i1 - s * r0;
        }
    }
}

template<int CB, int TB>
__device__ __forceinline__ void cnot_gate(float re[16], float im[16]) {
#pragma unroll
    for (int i = 0; i < 16; ++i) {
        if ((i & CB) && !(i & TB)) {
            const int j = i | TB;
            float t;
            t = re[i]; re[i] = re[j]; re[j] = t;
            t = im[i]; im[i] = im[j]; im[j] = t;
        }
    }
}

__global__ __launch_bounds__(256)
void fused_hybrid_kernel(const float* __restrict__ x,
                         const float* __restrict__ Wb,
                         const float* __restrict__ b1,
                         const float* __restrict__ qp,   // [L,4]
                         const float* __restrict__ W2,   // [1,4]
                         const float* __restrict__ b2,   // [1]
                         float* __restrict__ out,
                         int D, int L) {
    __shared__ float h_lds[256 * NQ];   // 4 KB: relu(x@W1.T + b1) per row

    const int tid  = threadIdx.x;
    const int wave = tid >> 5;
    const int lane = tid & 31;
    const int n    = lane & 15;     // column (N) index inside the 16-wide tile
    const int hb   = lane >> 4;     // half-wave select
    const int kOff = hb * 2;        // A/B operands: lanes 16-31 hold K+2/K+3

    // ---- Stage 1: h = relu(x @ W1.T + b1) via V_WMMA_F32_16X16X4_F32 ----
    const long rowBase = (long)blockIdx.x * 256 + (long)wave * 32;
    const float* px0 = x + (rowBase + n) * (long)D + kOff;        // rows 0..15
    const float* px1 = px0 + (long)16 * D;                        // rows 16..31
    const float* pwb = Wb + hb * 32 + n * 2;                      // B operand

    v8f c0 = {0.f, 0.f, 0.f, 0.f, 0.f, 0.f, 0.f, 0.f};
    v8f c1 = {0.f, 0.f, 0.f, 0.f, 0.f, 0.f, 0.f, 0.f};

    for (int k0 = 0; k0 < D; k0 += 4) {
        v2f a0 = *(const v2f*)(px0 + k0);
        v2f a1 = *(const v2f*)(px1 + k0);
        v2f bv = *(const v2f*)(pwb + k0 * 16);
        c0 = __builtin_amdgcn_wmma_f32_16x16x4_f32(
                 false, a0, false, bv, (short)0, c0, false, false);
        c1 = __builtin_amdgcn_wmma_f32_16x16x4_f32(
                 false, a1, false, bv, (short)0, c1, false, false);
    }

    // D layout: VGPR r, lanes 0-15 -> M=r, lanes 16-31 -> M=r+8; N = lane&15
    if (n < NQ) {
        const float bias = b1[n];
        const int r0 = wave * 32 + hb * 8;
#pragma unroll
        for (int r = 0; r < 8; ++r) {
            float v0 = c0[r] + bias;
            float v1 = c1[r] + bias;
            h_lds[(r0 + r) * NQ + n]      = v0 > 0.f ? v0 : 0.f;
            h_lds[(r0 + r + 16) * NQ + n] = v1 > 0.f ? v1 : 0.f;
        }
    }
    __syncthreads();

    // ---- Stage 2: 4-qubit circuit, one row per thread ----
    float re[16], im[16];
#pragma unroll
    for (int i = 0; i < 16; ++i) { re[i] = 0.f; im[i] = 0.f; }
    re[0] = 1.0f;

    float s, c;
    // AngleEmbedding: RX(h[w]) on wire w  (wire w -> bit 1<<(3-w))
    sincosf(0.5f * h_lds[tid * NQ + 0], &s, &c); rx_gate<8>(re, im, c, s);
    sincosf(0.5f * h_lds[tid * NQ + 1], &s, &c); rx_gate<4>(re, im, c, s);
    sincosf(0.5f * h_lds[tid * NQ + 2], &s, &c); rx_gate<2>(re, im, c, s);
    sincosf(0.5f * h_lds[tid * NQ + 3], &s, &c); rx_gate<1>(re, im, c, s);

    // BasicEntanglerLayers: RX per wire, then CNOT ring (0,1)(1,2)(2,3)(3,0)
    for (int l = 0; l < L; ++l) {
        const float* wl = qp + l * NQ;   // uniform -> scalar loads
        sincosf(0.5f * wl[0], &s, &c); rx_gate<8>(re, im, c, s);
        sincosf(0.5f * wl[1], &s, &c); rx_gate<4>(re, im, c, s);
        sincosf(0.5f * wl[2], &s, &c); rx_gate<2>(re, im, c, s);
        sincosf(0.5f * wl[3], &s, &c); rx_gate<1>(re, im, c, s);
        cnot_gate<8, 4>(re, im);
        cnot_gate<4, 2>(re, im);
        cnot_gate<2, 1>(re, im);
        cnot_gate<1, 8>(re, im);
    }

    // ---- Stage 3: out = sum_i |amp_i|^2 * (sum_w sign_w(i)*W2[w]) + b2 ----
    const float w20 = W2[0], w21 = W2[1], w22 = W2[2], w23 = W2[3];
    float acc = b2[0];
#pragma unroll
    for (int i = 0; i < 16; ++i) {
        float coef = ((i & 8) ? -w20 : w20) + ((i & 4) ? -w21 : w21)
                   + ((i & 2) ? -w22 : w22) + ((i & 1) ? -w23 : w23);
        acc += (re[i] * re[i] + im[i] * im[i]) * coef;
    }
    out[(long)blockIdx.x * 256 + tid] = acc;
}

extern "C" void kernel_launch(void* const* d_in, const int* in_sizes, int n_in,
                              void* d_out, int out_size, void* d_ws, size_t ws_size,
                              hipStream_t stream) {
    const float* x  = (const float*)d_in[0];   // [B, D]
    const float* W1 = (const float*)d_in[1];   // [4, D]
    const float* b1 = (const float*)d_in[2];   // [4]
    const float* qp = (const float*)d_in[3];   // [L, 4]
    const float* W2 = (const float*)d_in[4];   // [1, 4]
    const float* b2 = (const float*)d_in[5];   // [1]
    float* out = (float*)d_out;

    const int  D = in_sizes[1] / NQ;           // 512
    const long B = (long)in_sizes[0] / D;      // 262144
    const int  L = in_sizes[3] / NQ;           // 3

    float* Wb = (float*)d_ws;                  // D*16 floats = 32 KB

    const int prepN = D * 16;
    prep_wb_kernel<<<(prepN + 255) / 256, 256, 0, stream>>>(W1, Wb, D);

    fused_hybrid_kernel<<<(int)(B / 256), 256, 0, stream>>>(
        x, Wb, b1, qp, W2, b2, out, D, L);
}